// SelfMatching_9835475108578
// MI455X (gfx1250) — compile-verified
//
#include <hip/hip_runtime.h>
#include <math.h>

// Problem constants (match reference)
#define B_ 4
#define S_ 4096
#define D_ 128
#define NEG_BIG 1e30f

typedef float v2f __attribute__((ext_vector_type(2)));
typedef float v8f __attribute__((ext_vector_type(8)));

// One workgroup per batch: compute key-softmax weights w[j] and the single
// output row row[b,:] = sum_j w[j] * x[j,:]  (additive attention collapses:
// softmax over j is independent of query index i).
__global__ __launch_bounds__(1024)
void sm_rowweights_kernel(const float* __restrict__ inputs,
                          const int*   __restrict__ mask,
                          const float* __restrict__ kern,
                          float*       __restrict__ rowOut)
{
    __shared__ float lds_t[S_];        // s2 -> masked scores -> exp -> weights
    __shared__ float red[32];          // per-wave partials
    __shared__ float red2[32 * 16];    // WMMA cross-chunk partials
    __shared__ float scal;             // block max, then block sum

    const int tid  = threadIdx.x;      // 0..1023
    const int lane = tid & 31;
    const int wave = tid >> 5;         // 0..31 (wave32)
    const int b    = blockIdx.x;

    const float* X = inputs + (size_t)b * S_ * D_;

    // k1 = kernel[1, :, 0]; each lane owns 4 contiguous components
    const float4 kv = *(const float4*)(kern + D_ + 4 * lane);

    // ---- Phase 1: s2[j] = X[j,:].k1 ; t[j] = s2 - (1-mask)*NEG_BIG ----
    float wmax = -INFINITY;
    for (int r = 0; r < S_ / 32; ++r) {            // 128 rows per wave
        const int j = wave * (S_ / 32) + r;
        const float4 xv = *(const float4*)(X + (size_t)j * D_ + 4 * lane);
        float p = xv.x * kv.x + xv.y * kv.y + xv.z * kv.z + xv.w * kv.w;
#pragma unroll
        for (int off = 16; off > 0; off >>= 1)     // wave32 butterfly sum
            p += __shfl_xor(p, off, 32);
        if (lane == 0) {
            const float m = (float)mask[b * S_ + j];
            const float t = p - (1.0f - m) * NEG_BIG;
            lds_t[j] = t;
            wmax = fmaxf(wmax, t);
        }
    }
    if (lane == 0) red[wave] = wmax;
    __syncthreads();
    if (wave == 0) {
        float m = red[lane];
#pragma unroll
        for (int off = 16; off > 0; off >>= 1)
            m = fmaxf(m, __shfl_xor(m, off, 32));
        if (lane == 0) scal = m;
    }
    __syncthreads();
    const float bmax = scal;

    // ---- Phase 2: exp, sum, normalize ----
    float lsum = 0.0f;
#pragma unroll
    for (int q = 0; q < 4; ++q) {
        const int j = tid * 4 + q;
        const float e = expf(lds_t[j] - bmax);
        lds_t[j] = e;
        lsum += e;
    }
#pragma unroll
    for (int off = 16; off > 0; off >>= 1)
        lsum += __shfl_xor(lsum, off, 32);
    if (lane == 0) red[wave] = lsum;
    __syncthreads();
    if (wave == 0) {
        float s = red[lane];
#pragma unroll
        for (int off = 16; off > 0; off >>= 1)
            s += __shfl_xor(s, off, 32);
        if (lane == 0) scal = s;
    }
    __syncthreads();
    const float inv = 1.0f / scal;
#pragma unroll
    for (int q = 0; q < 4; ++q)
        lds_t[tid * 4 + q] *= inv;
    __syncthreads();

    // ---- Phase 3: row[d] = sum_j w[j]*x[j,d] via V_WMMA_F32_16X16X4_F32 ----
    // wave -> (d-tile of 16 columns, j-chunk of 1024). A = w replicated over
    // all 16 M-rows (A 16x4 layout: lanes 0-15 hold K=0,1; lanes 16-31 K=2,3).
    // B = 4x16 tile of X (mirrored layout). All D rows identical; acc[0] on
    // lane l holds the partial for column d0 + (l & 15). EXEC all-1s here.
    const int dt    = wave & 7;
    const int chunk = wave >> 3;
    const int d0    = dt * 16;
    const int lm    = lane & 15;
    const int hi    = lane >> 4;                   // 0: K=0,1  1: K=2,3

    v8f acc = {0.f, 0.f, 0.f, 0.f, 0.f, 0.f, 0.f, 0.f};
    const int jbase = chunk * (S_ / 4);            // 1024
    for (int it = 0; it < (S_ / 4) / 4; ++it) {    // 256 WMMA accumulations
        const int ja = jbase + it * 4 + hi * 2;
        v2f a, bv;
        a.x = lds_t[ja];
        a.y = lds_t[ja + 1];
        const float* xp = X + (size_t)ja * D_ + d0 + lm;
        bv.x = xp[0];
        bv.y = xp[D_];
        acc = __builtin_amdgcn_wmma_f32_16x16x4_f32(
                  /*neg_a=*/false, a, /*neg_b=*/false, bv,
                  /*c_mod=*/(short)0, acc, /*reuse_a=*/false, /*reuse_b=*/false);
    }
    if (hi == 0) red2[wave * 16 + lm] = acc[0];
    __syncthreads();
    if (tid < D_) {
        const int dtile = tid >> 4, l = tid & 15;
        float s = 0.0f;
#pragma unroll
        for (int c = 0; c < 4; ++c)
            s += red2[(c * 8 + dtile) * 16 + l];
        rowOut[b * D_ + tid] = s;
    }
}

// Broadcast: out[b,i,:] = row[b,:] for all i. Dominant HBM write (8 MB),
// float4 stores, row reads hit L2 (512 B per batch).
__global__ __launch_bounds__(256)
void sm_broadcast_kernel(const float* __restrict__ row,
                         float4*      __restrict__ out)
{
    const size_t idx = (size_t)blockIdx.x * 256 + threadIdx.x;  // float4 index
    const int    d4  = (int)(idx & (D_ / 4 - 1));               // 0..31
    const size_t bs  = idx >> 5;                                // b*S + s
    const int    b   = (int)(bs >> 12);                         // /S_
    out[idx] = ((const float4*)(row + b * D_))[d4];
}

extern "C" void kernel_launch(void* const* d_in, const int* in_sizes, int n_in,
                              void* d_out, int out_size, void* d_ws, size_t ws_size,
                              hipStream_t stream)
{
    (void)in_sizes; (void)n_in; (void)out_size; (void)ws_size;
    const float* inputs = (const float*)d_in[0];   // [B,S,D] f32
    const int*   mask   = (const int*)  d_in[1];   // [B,S]   i32
    const float* kern   = (const float*)d_in[2];   // [2,D,1] f32
    // d_in[3] = bias: constant per softmax row -> cancels exactly; unused.
    float* out = (float*)d_out;                    // [B,S,D] f32
    float* row = (float*)d_ws;                     // B*D floats scratch

    sm_rowweights_kernel<<<dim3(B_), dim3(1024), 0, stream>>>(inputs, mask, kern, row);

    const int n4 = B_ * S_ * D_ / 4;               // 524288 float4s
    sm_broadcast_kernel<<<dim3(n4 / 256), dim3(256), 0, stream>>>(row, (float4*)out);
}